// Model_19894288515658
// MI455X (gfx1250) — compile-verified
//
#include <hip/hip_runtime.h>
#include <math.h>

#define Bn 8
#define Sn 128
#define Dn 256
#define Hn 8
#define Tn 129
#define HDn 32
#define NEG_INF_F (-1000000000.0f)

typedef __attribute__((ext_vector_type(2))) float v2f;
typedef __attribute__((ext_vector_type(8))) float v8f;

// ---------------------------------------------------------------------------
// Wt[n*256+k] = W[k*256+n]  (one 256x256 matrix)
// ---------------------------------------------------------------------------
__global__ __launch_bounds__(256)
void transpose256(const float* __restrict__ W, float* __restrict__ Wt) {
  int idx = blockIdx.x * 256 + threadIdx.x;   // grid 256 -> 65536 threads
  int n = idx >> 8, k = idx & 255;
  Wt[idx] = W[(size_t)k * Dn + n];            // coalesced writes
}

// ---------------------------------------------------------------------------
// Y(nrows x 256) = X(nrows x 256) @ W(256 x 256) + bias, fp32 WMMA 16x16x4.
// W is pre-transposed (Wt[n,k]) so B fetches are contiguous b64 K-pairs.
// One wave computes a 16x64 tile: 4 accumulators share each A fetch.
// Double-buffered: next K-step's 5 loads are in flight during the 4 WMMAs.
// Row guard is a CLAMP (not a predicate) so EXEC stays all-ones in the loop.
// ---------------------------------------------------------------------------
__global__ __launch_bounds__(256)
void gemm256_wmma(const float* __restrict__ X, const float* __restrict__ Wt,
                  const float* __restrict__ bias, float* __restrict__ Y,
                  int nrows, int nwaves) {
  int wave = (blockIdx.x << 3) + (threadIdx.x >> 5);
  if (wave >= nwaves) return;                 // wave-uniform exit
  int lane   = threadIdx.x & 31;
  int tile_m = wave >> 2;                     // 4 column-groups of 64
  int cg     = wave & 3;
  int row0 = tile_m << 4;
  int col0 = cg << 6;
  int half = lane >> 4;                       // K-pair select per ISA layout
  int l    = lane & 15;
  int ar = row0 + l;
  if (ar >= nrows) ar = nrows - 1;            // clamped rows are never stored
  const v2f* xp  = (const v2f*)(X  + (size_t)ar * Dn);
  const v2f* wp0 = (const v2f*)(Wt + (size_t)(col0 +  0 + l) * Dn);
  const v2f* wp1 = (const v2f*)(Wt + (size_t)(col0 + 16 + l) * Dn);
  const v2f* wp2 = (const v2f*)(Wt + (size_t)(col0 + 32 + l) * Dn);
  const v2f* wp3 = (const v2f*)(Wt + (size_t)(col0 + 48 + l) * Dn);

  v8f acc0 = {0.f,0.f,0.f,0.f,0.f,0.f,0.f,0.f};
  v8f acc1 = acc0, acc2 = acc0, acc3 = acc0;

  v2f a  = xp[half];
  v2f b0 = wp0[half], b1 = wp1[half], b2 = wp2[half], b3 = wp3[half];

  #pragma unroll 4
  for (int k2 = 2; k2 < 128; k2 += 2) {       // k2 = K/2; K-step of 4
    v2f an  = xp [k2 + half];
    v2f bn0 = wp0[k2 + half];
    v2f bn1 = wp1[k2 + half];
    v2f bn2 = wp2[k2 + half];
    v2f bn3 = wp3[k2 + half];
    acc0 = __builtin_amdgcn_wmma_f32_16x16x4_f32(false, a, false, b0, (short)0, acc0, false, false);
    acc1 = __builtin_amdgcn_wmma_f32_16x16x4_f32(false, a, false, b1, (short)0, acc1, false, false);
    acc2 = __builtin_amdgcn_wmma_f32_16x16x4_f32(false, a, false, b2, (short)0, acc2, false, false);
    acc3 = __builtin_amdgcn_wmma_f32_16x16x4_f32(false, a, false, b3, (short)0, acc3, false, false);
    a = an; b0 = bn0; b1 = bn1; b2 = bn2; b3 = bn3;
  }
  acc0 = __builtin_amdgcn_wmma_f32_16x16x4_f32(false, a, false, b0, (short)0, acc0, false, false);
  acc1 = __builtin_amdgcn_wmma_f32_16x16x4_f32(false, a, false, b1, (short)0, acc1, false, false);
  acc2 = __builtin_amdgcn_wmma_f32_16x16x4_f32(false, a, false, b2, (short)0, acc2, false, false);
  acc3 = __builtin_amdgcn_wmma_f32_16x16x4_f32(false, a, false, b3, (short)0, acc3, false, false);

  float bs0 = bias ? bias[col0 +  0 + l] : 0.f;
  float bs1 = bias ? bias[col0 + 16 + l] : 0.f;
  float bs2 = bias ? bias[col0 + 32 + l] : 0.f;
  float bs3 = bias ? bias[col0 + 48 + l] : 0.f;
  #pragma unroll
  for (int r = 0; r < 8; ++r) {               // C/D: M = half*8 + r, N = l
    int row = row0 + half * 8 + r;
    if (row < nrows) {
      float* yp = Y + (size_t)row * Dn + col0 + l;
      yp[ 0] = acc0[r] + bs0;
      yp[16] = acc1[r] + bs1;
      yp[32] = acc2[r] + bs2;
      yp[48] = acc3[r] + bs3;
    }
  }
}

// qdot[bi*H+h] = Q[bi, h*32 .. h*32+31] . w[0..31]
__global__ __launch_bounds__(256)
void head_dot(const float* __restrict__ P, const float* __restrict__ w,
              float* __restrict__ out) {
  int idx = blockIdx.x * 256 + threadIdx.x;
  if (idx >= Bn * Tn * Hn) return;
  int h  = idx & 7;
  int bi = idx >> 3;
  const float* p = P + (size_t)bi * Dn + h * HDn;
  float s = 0.f;
  #pragma unroll
  for (int d = 0; d < HDn; ++d) s += p[d] * w[d];
  out[idx] = s;
}

// U[k*H+h] = sum_d We2[k, h*32+d] * wa_e[d];  eb[h] = be2[h*32..] . wa_e
__global__ __launch_bounds__(256)
void build_u(const float* __restrict__ We2, const float* __restrict__ be2,
             const float* __restrict__ wa_e, float* __restrict__ U,
             float* __restrict__ eb) {
  int idx = blockIdx.x * 256 + threadIdx.x;
  if (idx >= Dn * Hn) return;
  int h = idx & 7;
  int k = idx >> 3;
  float s = 0.f;
  #pragma unroll
  for (int d = 0; d < HDn; ++d) s += We2[(size_t)k * Dn + h * HDn + d] * wa_e[d];
  U[idx] = s;
  if (idx < Hn) {
    float e = 0.f;
    #pragma unroll
    for (int d = 0; d < HDn; ++d) e += be2[idx * HDn + d] * wa_e[d];
    eb[idx] = e;
  }
}

// One block per (b,i); one wave per j. x = A[ra] + Bm[j-1]; LN+ReLU; 8 head dots.
__global__ __launch_bounds__(256)
void edge_logits(const float* __restrict__ Abuf, const float* __restrict__ Bbuf,
                 const float* __restrict__ ln_g, const float* __restrict__ ln_b,
                 const float* __restrict__ U, const float* __restrict__ eb,
                 const float* __restrict__ qdot, const float* __restrict__ kdot,
                 const float* __restrict__ prior, const float* __restrict__ ba,
                 float* __restrict__ logits) {
  int b = blockIdx.x / Tn;
  int i = blockIdx.x % Tn;
  int wave = threadIdx.x >> 5, lane = threadIdx.x & 31;
  float bav = ba[0];
  for (int j = wave; j < Tn; j += 8) {
    if (j == 0 || i == j) {                      // adj == 0 -> masked
      if (lane < Hn)
        logits[(((size_t)(b * Hn + lane) * Tn) + i) * Tn + j] = NEG_INF_F;
      continue;
    }
    int ra = (i == 0) ? (j - 1) : (i - 1);       // cls row reuses desc_j
    const float* Ap = Abuf + ((size_t)b * Sn + ra) * Dn;     // has be1 folded in
    const float* Bp = Bbuf + ((size_t)b * Sn + (j - 1)) * Dn;
    float x[8];
    float sum = 0.f;
    #pragma unroll
    for (int t = 0; t < 8; ++t) {
      int k = lane + (t << 5);
      float xv = Ap[k] + Bp[k];
      x[t] = xv; sum += xv;
    }
    #pragma unroll
    for (int off = 16; off; off >>= 1) sum += __shfl_xor(sum, off, 32);
    float mu = sum * (1.0f / Dn);
    float vs = 0.f;
    #pragma unroll
    for (int t = 0; t < 8; ++t) { float d = x[t] - mu; vs += d * d; }
    #pragma unroll
    for (int off = 16; off; off >>= 1) vs += __shfl_xor(vs, off, 32);
    float rstd = rsqrtf(vs * (1.0f / Dn) + 1e-5f);
    float eacc[8] = {0.f,0.f,0.f,0.f,0.f,0.f,0.f,0.f};
    #pragma unroll
    for (int t = 0; t < 8; ++t) {
      int k = lane + (t << 5);
      float r = (x[t] - mu) * rstd * ln_g[k] + ln_b[k];
      r = fmaxf(r, 0.f);
      const float* up = U + (size_t)k * Hn;
      #pragma unroll
      for (int h = 0; h < 8; ++h) eacc[h] += r * up[h];
    }
    #pragma unroll
    for (int h = 0; h < 8; ++h) {
      #pragma unroll
      for (int off = 16; off; off >>= 1) eacc[h] += __shfl_xor(eacc[h], off, 32);
    }
    if (lane < Hn) {
      int h = lane;
      float lg = qdot[(b * Tn + i) * Hn + h] + kdot[(b * Tn + j) * Hn + h] +
                 eacc[h] + eb[h] + bav;
      if (i >= 1) {                              // prior logit bias (i,j >= 1)
        float p = prior[(((size_t)(b * Hn + h) * Sn) + (i - 1)) * Sn + (j - 1)];
        p = fminf(fmaxf(p, 1e-6f), 1.0f - 1e-6f);
        lg += logf(p) - logf(1.0f - p);
      }
      logits[(((size_t)(b * Hn + h) * Tn) + i) * Tn + j] = lg;
    }
  }
}

// In-place row softmax over j (T=129), one wave per (b,h,i) row.
__global__ __launch_bounds__(256)
void softmax_rows(float* __restrict__ attn) {
  int wave = threadIdx.x >> 5, lane = threadIdx.x & 31;
  int row = blockIdx.x * 8 + wave;
  if (row >= Bn * Hn * Tn) return;
  float* p = attn + (size_t)row * Tn;
  float v[5];
  int n = 0;
  float mx = -3.0e38f;
  for (int j = lane; j < Tn; j += 32) { float x = p[j]; v[n++] = x; mx = fmaxf(mx, x); }
  #pragma unroll
  for (int off = 16; off; off >>= 1) mx = fmaxf(mx, __shfl_xor(mx, off, 32));
  float s = 0.f;
  for (int t = 0; t < n; ++t) { v[t] = expf(v[t] - mx); s += v[t]; }
  #pragma unroll
  for (int off = 16; off; off >>= 1) s += __shfl_xor(s, off, 32);
  float inv = 1.0f / s;
  n = 0;
  for (int j = lane; j < Tn; j += 32) p[j] = v[n++] * inv;
}

// ctx[b,i,h,d] = sum_j attn[b,h,i,j] * V[b,j,h*32+d]  (output already (B,T,H,HD))
__global__ __launch_bounds__(256)
void ctx_gemv(const float* __restrict__ attn, const float* __restrict__ V,
              float* __restrict__ out) {
  int idx = blockIdx.x * 256 + threadIdx.x;
  if (idx >= Bn * Tn * Hn * HDn) return;
  int d  = idx & 31;
  int h  = (idx >> 5) & 7;
  int bi = idx >> 8;                       // b*T + i
  int b = bi / Tn, i = bi % Tn;
  const float* arow = attn + (((size_t)(b * Hn + h) * Tn) + i) * Tn;
  const float* vp   = V + (size_t)b * Tn * Dn + h * HDn + d;
  float s = 0.f;
  for (int j = 0; j < Tn; ++j) s += arow[j] * vp[(size_t)j * Dn];
  out[idx] = s;
}

extern "C" void kernel_launch(void* const* d_in, const int* in_sizes, int n_in,
                              void* d_out, int out_size, void* d_ws, size_t ws_size,
                              hipStream_t stream) {
  const float* desc  = (const float*)d_in[0];
  const float* nve   = (const float*)d_in[1];
  const float* prior = (const float*)d_in[2];
  const float* Wq  = (const float*)d_in[3];
  const float* bq  = (const float*)d_in[4];
  const float* Wk  = (const float*)d_in[5];
  const float* bk  = (const float*)d_in[6];
  const float* Wv  = (const float*)d_in[7];
  const float* bv  = (const float*)d_in[8];
  const float* wa  = (const float*)d_in[9];
  const float* ba  = (const float*)d_in[10];
  const float* We1 = (const float*)d_in[11];
  const float* be1 = (const float*)d_in[12];
  const float* lng = (const float*)d_in[13];
  const float* lnb = (const float*)d_in[14];
  const float* We2 = (const float*)d_in[15];
  const float* be2 = (const float*)d_in[16];

  const int BT = Bn * Tn;   // 1032
  const int BS = Bn * Sn;   // 1024
  const size_t MM = (size_t)Dn * Dn;   // 65536

  float* ws   = (float*)d_ws;
  float* Q    = ws;  ws += (size_t)BT * Dn;
  float* K    = ws;  ws += (size_t)BT * Dn;
  float* V    = ws;  ws += (size_t)BT * Dn;
  float* Abuf = ws;  ws += (size_t)BS * Dn;
  float* Bbuf = ws;  ws += (size_t)BS * Dn;
  float* qdot = ws;  ws += BT * Hn;
  float* kdot = ws;  ws += BT * Hn;
  float* U    = ws;  ws += Dn * Hn;
  float* eb   = ws;  ws += Hn;
  float* WqT  = ws;  ws += MM;
  float* WkT  = ws;  ws += MM;
  float* WvT  = ws;  ws += MM;
  float* WaT  = ws;  ws += MM;   // We1 top half, transposed
  float* WbT  = ws;  ws += MM;   // We1 bottom half, transposed

  float* ctx_out  = (float*)d_out;
  float* attn_out = ctx_out + (size_t)BT * Hn * HDn;   // attn lives in d_out

  // Pre-transpose weights so GEMM B-operand fetches are contiguous b64 pairs.
  transpose256<<<256, 256, 0, stream>>>(Wq, WqT);
  transpose256<<<256, 256, 0, stream>>>(Wk, WkT);
  transpose256<<<256, 256, 0, stream>>>(Wv, WvT);
  transpose256<<<256, 256, 0, stream>>>(We1, WaT);
  transpose256<<<256, 256, 0, stream>>>(We1 + MM, WbT);

  const int w1032 = ((BT + 15) / 16) * 4;   // 65 row-tiles * 4 col-groups = 260
  const int w1024 = ((BS + 15) / 16) * 4;   // 256

  gemm256_wmma<<<(w1032 + 7) / 8, 256, 0, stream>>>(nve, WqT, bq, Q, BT, w1032);
  gemm256_wmma<<<(w1032 + 7) / 8, 256, 0, stream>>>(nve, WkT, bk, K, BT, w1032);
  gemm256_wmma<<<(w1032 + 7) / 8, 256, 0, stream>>>(nve, WvT, bv, V, BT, w1032);
  gemm256_wmma<<<(w1024 + 7) / 8, 256, 0, stream>>>(desc, WaT, be1, Abuf, BS, w1024);
  gemm256_wmma<<<(w1024 + 7) / 8, 256, 0, stream>>>(desc, WbT, nullptr, Bbuf, BS, w1024);

  head_dot<<<(BT * Hn + 255) / 256, 256, 0, stream>>>(Q, wa, qdot);
  head_dot<<<(BT * Hn + 255) / 256, 256, 0, stream>>>(K, wa + HDn, kdot);
  build_u<<<(Dn * Hn + 255) / 256, 256, 0, stream>>>(We2, be2, wa + 2 * HDn, U, eb);

  edge_logits<<<BT, 256, 0, stream>>>(Abuf, Bbuf, lng, lnb, U, eb,
                                      qdot, kdot, prior, ba, attn_out);
  softmax_rows<<<(Bn * Hn * Tn + 7) / 8, 256, 0, stream>>>(attn_out);
  ctx_gemv<<<(BT * Hn * HDn + 255) / 256, 256, 0, stream>>>(attn_out, V, ctx_out);
}